// EntityAggregator_5849745457882
// MI455X (gfx1250) — compile-verified
//
#include <hip/hip_runtime.h>
#include <hip/hip_bf16.h>

typedef __attribute__((ext_vector_type(2))) float v2f;
typedef __attribute__((ext_vector_type(8))) float v8f;

// Problem constants (from reference)
constexpr int kB = 2, kT = 4, kV = 4, kH = 128, kW = 128;
constexpr int BT   = kB * kT;            // 8 batch*time slices
constexpr int P    = kV * kH * kW;       // 65536 pixels per (b,t)
constexpr int NQ   = 64;
constexpr int QDIM = 256;
constexpr int IDIM = 32;
constexpr int MDIM = 16;
constexpr int DTOT = IDIM + MDIM;        // 48 combined output dims
constexpr long long TOTPIX   = (long long)BT * P;          // 524288
constexpr long long OUT_INST = TOTPIX;                     // after entity ids
constexpr long long OUT_MOT  = TOTPIX + TOTPIX * IDIM;     // after inst

__device__ __forceinline__ v8f wmma_f32(v2f a, v2f b, v8f c) {
  return __builtin_amdgcn_wmma_f32_16x16x4_f32(
      /*neg_a=*/false, a, /*neg_b=*/false, b,
      /*c_mod=*/(short)0, c, /*reuse_a=*/false, /*reuse_b=*/false);
}

// ---------------------------------------------------------------------------
// Kernel 1: query projection  qc[bt][m][0:48] = qf[bt,m,:] @ [W_inst|W_motion] + bias
// grid = BT, block = 384 (12 waves). Wave w owns (mt = w/3, nt = w%3).
// ---------------------------------------------------------------------------
__global__ __launch_bounds__(384)
void qproj_kernel(const float* __restrict__ qf,
                  const float* __restrict__ Wi, const float* __restrict__ bi,
                  const float* __restrict__ Wm, const float* __restrict__ bm,
                  float* __restrict__ qc) {
  const int bt   = blockIdx.x;
  const int wave = threadIdx.x >> 5;
  const int lane = threadIdx.x & 31;
  const int mt   = wave / 3;           // 0..3 -> rows of query matrix
  const int nt   = wave % 3;           // 0,1 -> W_inst halves, 2 -> W_motion
  const int hi   = lane >> 4;
  const int li   = lane & 15;

  const int m = mt * 16 + li;          // A-matrix row for this lane
  const float* qrow = qf + ((long long)bt * NQ + m) * QDIM;

  const float* Wp;
  int nstride, noff;
  if (nt < 2) { Wp = Wi; nstride = IDIM; noff = nt * 16; }
  else        { Wp = Wm; nstride = MDIM; noff = 0; }

  v8f acc = {};
#pragma unroll 4
  for (int k0 = 0; k0 < QDIM; k0 += 4) {
    const int kr = k0 + 2 * hi;        // A lane layout: VGPR0/1 hold K = 2*hi, 2*hi+1
    v2f a; a.x = qrow[kr];             a.y = qrow[kr + 1];
    v2f b; b.x = Wp[kr * nstride + noff + li];
           b.y = Wp[(kr + 1) * nstride + noff + li];
    acc = wmma_f32(a, b, acc);
  }

  const float bias = (nt < 2) ? bi[noff + li] : bm[li];
  const int   ccol = (nt < 2) ? (nt * 16 + li) : (IDIM + li);
#pragma unroll
  for (int r = 0; r < 8; ++r) {
    const int row = mt * 16 + r + hi * 8;   // C/D layout: VGPR r -> M = r + 8*hi
    qc[((long long)bt * NQ + row) * DTOT + ccol] = acc[r] + bias;
  }
}

// ---------------------------------------------------------------------------
// Kernel 2: main bandwidth-bound pass.
// grid = BT * 128 blocks; block = 256 threads (8 waves).
// Each block: one (b,t), 512 contiguous pixels. Each wave: 4 tiles of 16 pixels.
// Per tile: pooled = fg_prob(16x64) @ qc(64x48) via 48x v_wmma_f32_16x16x4_f32.
// K is permuted: WMMA (step s, slot j) <- fg column 32*(j>>1) + 2s + (j&1),
// so each lane streams a contiguous 32-float prob run and each B fragment is
// one aligned ds_load_b64 from a pair-interleaved LDS layout.
// ---------------------------------------------------------------------------
__global__ __launch_bounds__(256)
void entity_main_kernel(const float* __restrict__ aff,
                        const float* __restrict__ mot,
                        const float* __restrict__ prob,
                        const int*   __restrict__ aq,
                        const float* __restrict__ qc,
                        float* __restrict__ out) {
  // Pair-interleaved query matrix: qs[(e*48 + c)*2 + (row&1)], e = row/2 (0..31)
  __shared__ float qs[NQ * DTOT];            // 12 KB

  const int bt    = blockIdx.x >> 7;
  const int chunk = blockIdx.x & 127;

  // Stage qc[bt] into LDS: b128 global loads, scatter into pair-interleaved layout.
  {
    const float4* src = (const float4*)(qc + (long long)bt * NQ * DTOT);
    for (int i = threadIdx.x; i < NQ * DTOT / 4; i += 256) {
      const float4 v = src[i];
      const int row = i / (DTOT / 4);
      const int c0  = (i % (DTOT / 4)) * 4;
      const int e   = row >> 1, r = row & 1;
      qs[(e * DTOT + c0 + 0) * 2 + r] = v.x;
      qs[(e * DTOT + c0 + 1) * 2 + r] = v.y;
      qs[(e * DTOT + c0 + 2) * 2 + r] = v.z;
      qs[(e * DTOT + c0 + 3) * 2 + r] = v.w;
    }
  }
  __syncthreads();

  const int wave = threadIdx.x >> 5;
  const int lane = threadIdx.x & 31;
  const int hi   = lane >> 4;
  const int li   = lane & 15;
  const long long btbase = (long long)bt * P;

  for (int t = 0; t < 4; ++t) {
    const int p0 = chunk * 512 + wave * 64 + t * 16;   // first pixel of tile
    const float* pr = prob + (btbase + p0 + li) * (NQ + 1);

    // Prefetch next tile's prob row for this lane (global_prefetch_b8).
    if (t < 3) __builtin_prefetch(pr + 16 * (NQ + 1), 0, 0);

    // Stream this lane's contiguous half of the fg-prob row + partial sum.
    const float bg = __builtin_nontemporal_load(pr);
    float avals[32];
    float lsum = 0.f;
#pragma unroll
    for (int j = 0; j < 32; ++j) {
      const float x = __builtin_nontemporal_load(pr + 1 + 32 * hi + j);
      avals[j] = x;
      lsum += x;
    }
    const float rowsum = lsum + __shfl_xor(lsum, 16, 32);    // combine K-halves
    const float scale  = (1.0f - bg) / fmaxf(rowsum, 1e-6f); // gate / fg_norm

    v8f ci0 = {}, ci1 = {}, cm = {};
#pragma unroll
    for (int s = 0; s < 16; ++s) {
      v2f a; a.x = avals[2 * s]; a.y = avals[2 * s + 1];
      const int e = 16 * hi + s;                       // B rows 32*hi+2s, +1 (pair)
      const v2f b0 = *(const v2f*)(qs + (e * DTOT + li) * 2);
      const v2f b1 = *(const v2f*)(qs + (e * DTOT + 16 + li) * 2);
      const v2f b2 = *(const v2f*)(qs + (e * DTOT + 32 + li) * 2);
      ci0 = wmma_f32(a, b0, ci0);
      ci1 = wmma_f32(a, b1, ci1);
      cm  = wmma_f32(a, b2, cm);
    }

    // Epilogue: C VGPR r -> pixel (r + 8*hi), col = li. Half-wave writes 64B runs.
#pragma unroll
    for (int r = 0; r < 8; ++r) {
      const int q = r + hi * 8;
      const float sc = __shfl(scale, q, 32);           // scale for this row's pixel
      const long long gp = btbase + p0 + q;
      const float* ai = aff + gp * IDIM;
      float*       oi = out + OUT_INST + gp * IDIM;
      __builtin_nontemporal_store(
          __builtin_nontemporal_load(ai + li) + sc * ci0[r], oi + li);
      __builtin_nontemporal_store(
          __builtin_nontemporal_load(ai + 16 + li) + sc * ci1[r], oi + 16 + li);
      __builtin_nontemporal_store(
          __builtin_nontemporal_load(mot + gp * MDIM + li) + sc * cm[r],
          out + OUT_MOT + gp * MDIM + li);
    }

    // entity_id passthrough (as float), one lane per pixel.
    if (hi == 0) {
      const long long gp = btbase + p0 + li;
      __builtin_nontemporal_store((float)__builtin_nontemporal_load(aq + gp),
                                  out + gp);
    }
  }
}

extern "C" void kernel_launch(void* const* d_in, const int* in_sizes, int n_in,
                              void* d_out, int out_size, void* d_ws, size_t ws_size,
                              hipStream_t stream) {
  const float* aff = (const float*)d_in[0];   // instance_affinity (B,T,V,H,W,32)
  const float* mot = (const float*)d_in[1];   // motion_code       (B,T,V,H,W,16)
  const float* qf  = (const float*)d_in[2];   // query_feat        (B,T,64,256)
  const float* pb  = (const float*)d_in[3];   // assignment_prob   (B,T,V,H,W,65)
  const int*   aq  = (const int*)  d_in[4];   // assigned_query    (B,T,V,H,W)
  const float* Wi  = (const float*)d_in[5];   // W_inst (256,32)
  const float* bi  = (const float*)d_in[6];   // b_inst (32)
  const float* Wm  = (const float*)d_in[7];   // W_motion (256,16)
  const float* bm  = (const float*)d_in[8];   // b_motion (16)
  float* out = (float*)d_out;
  float* qc  = (float*)d_ws;                  // (BT, NQ, 48) = 96 KB

  qproj_kernel<<<dim3(BT), dim3(384), 0, stream>>>(qf, Wi, bi, Wm, bm, qc);
  entity_main_kernel<<<dim3(BT * 128), dim3(256), 0, stream>>>(aff, mot, pb, aq, qc, out);
}